// SamplingFeatures_970662609239
// MI455X (gfx1250) — compile-verified
//
#include <hip/hip_runtime.h>

// ---------------------------------------------------------------------------
// SamplingFeatures: radial nearest-neighbor feature sampling.
// B=8, C=32, N_RAYS=32, H=W=256, nd_sampling=1 (channel == ray index).
//
// Pure HBM-bandwidth-bound (~470 MB traffic, ~0.5 flop/byte): no contraction
// dim -> WMMA inapplicable. MI455X optimization targets:
//   * 4 px/thread, b128 vector loads/stores, fully coalesced per wave32
//   * non-temporal stores for 335MB of write-once output (+ NT dist load),
//     preserving 192MB-L2 residency of the 64MB gathered feature map
//   * global_prefetch_b8 on the gather row (CDNA5 prefetch path)
//   * block-uniform (b,ray) decoded from blockIdx -> SGPR address bases
//   * v_sin/v_cos with exact period-1 argument r/32 (no libm arg reduction)
//   * v_rndne_f32 rounding == torch nearbyint (round half to even)
// ---------------------------------------------------------------------------

typedef float v4f __attribute__((ext_vector_type(4)));

namespace {
constexpr int Bn = 8;
constexpr int NR = 32;
constexpr int Hn = 256;
constexpr int Wn = 256;
constexpr int HW = Hn * Wn;
// output segment offsets (floats), in reference return order
constexpr long long OUT_GRID = (long long)Bn * NR * HW;                // 16,777,216
constexpr long long OUT_OFFS = OUT_GRID + (long long)NR * Bn * HW * 2; // 50,331,648
} // namespace

__global__ __launch_bounds__(256) void SamplingFeatures_kernel(
    const float* __restrict__ fm,    // (B, C=32, H, W)
    const float* __restrict__ dist,  // (B, NR=32, H, W)
    float* __restrict__ out)
{
  // ---- block-uniform decode (stays in SGPRs) ----
  // grid.x = 16384 blocks of 256 threads; x-quad fastest.
  const int br = blockIdx.x >> 6;                       // 0..255 == b*32 + r (uniform)
  const int r  = br & (NR - 1);                         // uniform
  const int b  = br >> 5;                               // uniform

  // per-ray direction: angle = r/32 revolutions (exact fp32). v_sin/v_cos
  // have period 1 -> no argument reduction, hardware-accurate.
  const float t  = (float)r * (1.0f / (float)NR);
  const float sn = __builtin_amdgcn_sinf(t);
  const float cs = __builtin_amdgcn_cosf(t);

  // ---- per-thread decode ----
  const int y  = ((blockIdx.x << 2) | (threadIdx.x >> 6)) & (Hn - 1);
  const int xq = threadIdx.x & (Wn / 4 - 1);            // 0..63
  const int x4 = xq << 2;

  const long long planeBase = (long long)br * HW;       // dist/sampled/fm(ch=r) share it
  const long long rowBase   = planeBase + y * Wn + x4;
  const float* __restrict__ fmPlane = fm + planeBase;

  // CDNA5 prefetch of the gather neighborhood (radius <= 16 px around (y,x))
  __builtin_prefetch(fmPlane + y * Wn + x4, 0, 3);

  // streamed read-once distances: non-temporal b128 load
  const v4f d4 = __builtin_nontemporal_load((const v4f*)(dist + rowBase));

  const float fy  = (float)y;
  const float inv = 2.0f / (float)(Wn - 1);             // W==H -> same scale both axes

  float nx[4], ny[4], sv[4];
#pragma unroll
  for (int i = 0; i < 4; ++i) {
    const float fx = (float)(x4 + i);
    const float d  = d4[i];
    const float ox = fmaf(cs, d, fx);
    const float oy = fmaf(sn, d, fy);
    const float nxi = fmaf(ox, inv, -1.0f);             // normalized coords
    const float nyi = fmaf(oy, inv, -1.0f);
    nx[i] = nxi;
    ny[i] = nyi;
    // grid_sample unnormalize (align_corners=False): ix = (nx+1)*W/2 - 0.5
    const float ixf = fmaf(nxi + 1.0f, (float)Wn * 0.5f, -0.5f);
    const float iyf = fmaf(nyi + 1.0f, (float)Hn * 0.5f, -0.5f);
    const int ix = (int)rintf(ixf);                     // v_rndne_f32
    const int iy = (int)rintf(iyf);
    const bool valid = ((unsigned)ix < (unsigned)Wn) & ((unsigned)iy < (unsigned)Hn);
    const int ixc = min(max(ix, 0), Wn - 1);
    const int iyc = min(max(iy, 0), Hn - 1);
    const float v = fmPlane[iyc * Wn + ixc];            // cached gather, L2-resident
    sv[i] = valid ? v : 0.0f;
  }

  // --- sampled: (B, NR, H, W), same flat layout as dist ---
  {
    v4f s4 = {sv[0], sv[1], sv[2], sv[3]};
    __builtin_nontemporal_store(s4, (v4f*)(out + rowBase));
  }

  // --- grid: (NR*B, H, W, 2), plane index r*B+b, interleaved (x,y) ---
  {
    const long long gBase =
        OUT_GRID + (((long long)(r * Bn + b) * Hn + y) * Wn + x4) * 2;
    v4f g0 = {nx[0], ny[0], nx[1], ny[1]};
    v4f g1 = {nx[2], ny[2], nx[3], ny[3]};
    __builtin_nontemporal_store(g0, (v4f*)(out + gBase));
    __builtin_nontemporal_store(g1, (v4f*)(out + gBase + 4));
  }

  // --- offsets_norm: (B, NR, 2, H, W): two contiguous planes per (b,r) ---
  {
    const long long oBase = OUT_OFFS + (long long)br * 2 * HW + y * Wn + x4;
    v4f ox4 = {nx[0], nx[1], nx[2], nx[3]};
    v4f oy4 = {ny[0], ny[1], ny[2], ny[3]};
    __builtin_nontemporal_store(ox4, (v4f*)(out + oBase));
    __builtin_nontemporal_store(oy4, (v4f*)(out + oBase + HW));
  }
}

extern "C" void kernel_launch(void* const* d_in, const int* in_sizes, int n_in,
                              void* d_out, int out_size, void* d_ws, size_t ws_size,
                              hipStream_t stream) {
  (void)in_sizes; (void)n_in; (void)out_size; (void)d_ws; (void)ws_size;
  const float* fm   = (const float*)d_in[0];  // feature_map (8,32,256,256) f32
  const float* dist = (const float*)d_in[1];  // dist        (8,32,256,256) f32
  // d_in[2] = nd_sampling scalar (==1), baked into the layout math.
  float* out = (float*)d_out;

  // 8*32*256*256 pixels / 4 px-per-thread / 256 threads = 16384 blocks;
  // 64 consecutive blocks share one (b,ray) plane (block-uniform scalars).
  const int threads = 256;                           // 8 wave32s per block
  const int blocks  = (Bn * NR * HW / 4) / threads;  // 16384
  SamplingFeatures_kernel<<<blocks, threads, 0, stream>>>(fm, dist, out);
}